// HingeLoss_2078764171740
// MI455X (gfx1250) — compile-verified
//
#include <hip/hip_runtime.h>
#include <hip/hip_bf16.h>

typedef __attribute__((ext_vector_type(2))) float v2f;
typedef __attribute__((ext_vector_type(4))) float v4f;
typedef __attribute__((ext_vector_type(8))) float v8f;

// ---------------------------------------------------------------------------
// Wave32 reduction using V_WMMA_F32_16X16X4_F32.
// A = 16x4 ones, B = 4x16 with the 32 lane values in B's first VGPR and zeros
// in the second. D[m][n] = sum_k B[k][n], so D row 0 (c[0]; lanes 0..15 hold
// N=0..15, lanes 16..31 replicate row 8 == row 0) holds the 16 column sums.
// Four xor-shuffles over the 16-lane group leave the 32-lane total in every
// lane. Requires EXEC == all ones (callers guarantee full waves).
// ---------------------------------------------------------------------------
__device__ __forceinline__ float wave_reduce_wmma(float v) {
    v2f a; a.x = 1.0f; a.y = 1.0f;   // A: 16x4 all-ones
    v2f b; b.x = v;    b.y = 0.0f;   // B: 32 lane values + 32 zeros
    v8f c = {};
    c = __builtin_amdgcn_wmma_f32_16x16x4_f32(
        /*neg_a=*/false, a, /*neg_b=*/false, b,
        /*c_mod=*/(short)0, c, /*reuse_a=*/false, /*reuse_b=*/false);
    float s = c[0];
    s += __shfl_xor(s, 1, 32);
    s += __shfl_xor(s, 2, 32);
    s += __shfl_xor(s, 4, 32);
    s += __shfl_xor(s, 8, 32);
    return s;  // total of all 32 lanes, in every lane
}

// Block reduction (blockDim.x == 256 -> 8 wave32s). All waves redundantly do
// the second stage so every WMMA executes with EXEC all-ones. Result valid in
// every thread. Caller must __syncthreads() before reusing `lds`.
__device__ __forceinline__ float block_reduce(float v, float* lds) {
    float w = wave_reduce_wmma(v);
    const int lane = threadIdx.x & 31;
    const int wid  = threadIdx.x >> 5;
    if (lane == 0) lds[wid] = w;
    __syncthreads();
    const int nw = blockDim.x >> 5;
    float s = (lane < nw) ? lds[lane] : 0.0f;
    return wave_reduce_wmma(s);
}

__device__ __forceinline__ float relu4_sum(v4f x) {
    float r0 = fmaxf(x.x + 25.0f, 0.0f);
    float r1 = fmaxf(x.y + 25.0f, 0.0f);
    float r2 = fmaxf(x.z + 25.0f, 0.0f);
    float r3 = fmaxf(x.w + 25.0f, 0.0f);
    return (r0 + r1) + (r2 + r3);
}

// ---------------------------------------------------------------------------
// Kernel 1: streaming reduction of relu(x + 25) over the whole tensor.
// Non-temporal b128 loads (stream > 192 MB L2, never cache it), unrolled x4
// so each wave keeps 4 independent global_load_b128 in flight (MLP to cover
// HBM latency at 23.3 TB/s). Fixed tree reduction => deterministic replays.
// ---------------------------------------------------------------------------
__global__ __launch_bounds__(256) void hinge_partial_kernel(
    const float* __restrict__ in, size_t nvec4, size_t ntail,
    float* __restrict__ partials) {
    const v4f* __restrict__ vin = (const v4f*)in;
    const size_t idx    = (size_t)blockIdx.x * blockDim.x + threadIdx.x;
    const size_t stride = (size_t)gridDim.x * blockDim.x;

    float s0 = 0.0f, s1 = 0.0f, s2 = 0.0f, s3 = 0.0f;
    size_t i = idx;
    // Unrolled-by-4: independent loads + independent accumulator chains.
    for (; i + 3 * stride < nvec4; i += 4 * stride) {
        v4f x0 = __builtin_nontemporal_load(vin + i);
        v4f x1 = __builtin_nontemporal_load(vin + i + stride);
        v4f x2 = __builtin_nontemporal_load(vin + i + 2 * stride);
        v4f x3 = __builtin_nontemporal_load(vin + i + 3 * stride);
        s0 += relu4_sum(x0);
        s1 += relu4_sum(x1);
        s2 += relu4_sum(x2);
        s3 += relu4_sum(x3);
    }
    for (; i < nvec4; i += stride) {
        v4f x = __builtin_nontemporal_load(vin + i);
        s0 += relu4_sum(x);
    }
    float v = (s0 + s1) + (s2 + s3);
    if (idx < ntail) {  // at most 3 scalar leftovers
        float x = in[nvec4 * 4 + idx];
        v += fmaxf(x + 25.0f, 0.0f);
    }

    __shared__ float lds[8];
    float s = block_reduce(v, lds);
    if (threadIdx.x == 0) partials[blockIdx.x] = s;
}

// ---------------------------------------------------------------------------
// Kernel 2: per-row target corrections.
//   corrAB[0] = sum_r relu(inputs[r, t_r] + 25)   (subtracted from both sums)
//   corrAB[1] = sum_r relu(50 - inputs[r, t_r])   (added to loss_m only)
// ---------------------------------------------------------------------------
__global__ __launch_bounds__(256) void hinge_corr_kernel(
    const float* __restrict__ in, const int* __restrict__ tgt,
    int n, int c, float* __restrict__ corrAB) {
    float a = 0.0f, b = 0.0f;
    for (int r = threadIdx.x; r < n; r += blockDim.x) {
        int t = tgt[r];
        float x = in[(size_t)r * (size_t)c + (size_t)t];
        a += fmaxf(x + 25.0f, 0.0f);
        b += fmaxf(50.0f - x, 0.0f);
    }
    __shared__ float lds[8];
    float sa = block_reduce(a, lds);
    __syncthreads();
    float sb = block_reduce(b, lds);
    if (threadIdx.x == 0) {
        corrAB[0] = sa;
        corrAB[1] = sb;
    }
}

// ---------------------------------------------------------------------------
// Kernel 3: combine block partials + corrections, write the two outputs.
// ---------------------------------------------------------------------------
__global__ __launch_bounds__(256) void hinge_final_kernel(
    const float* __restrict__ partials, int nb,
    const float* __restrict__ corrAB, float* __restrict__ out, float inv_n) {
    float v = 0.0f;
    for (int i = threadIdx.x; i < nb; i += blockDim.x) v += partials[i];
    __shared__ float lds[8];
    float loss_all = block_reduce(v, lds);
    if (threadIdx.x == 0) {
        float A = corrAB[0];
        float B = corrAB[1];
        float loss_inter = loss_all - A;
        float loss_m     = loss_inter + B;
        out[0] = loss_m * inv_n;      // reference return order: (loss_m, loss_inter)
        out[1] = loss_inter * inv_n;
    }
}

extern "C" void kernel_launch(void* const* d_in, const int* in_sizes, int n_in,
                              void* d_out, int out_size, void* d_ws, size_t ws_size,
                              hipStream_t stream) {
    const float* in  = (const float*)d_in[0];
    const int*   tgt = (const int*)d_in[1];
    float*       out = (float*)d_out;
    float*       ws  = (float*)d_ws;

    const size_t total = (size_t)in_sizes[0];   // n * c
    const int    n     = in_sizes[1];
    const int    c     = (int)(total / (size_t)n);

    // Block-partials budget: 2048 blocks x 8 wave32s, each wave with 4
    // in-flight b128 loads -> plenty of concurrency for 23.3 TB/s HBM.
    int NB = 2048;
    const size_t need = ((size_t)NB + 2) * sizeof(float);
    if (ws_size < need) {
        long avail = (long)(ws_size / sizeof(float)) - 2;
        NB = (avail < 1) ? 1 : (int)avail;
        if (NB > 2048) NB = 2048;
    }

    const size_t nvec4 = total / 4;
    const size_t ntail = total - nvec4 * 4;

    hinge_partial_kernel<<<NB, 256, 0, stream>>>(in, nvec4, ntail, ws);
    hinge_corr_kernel<<<1, 256, 0, stream>>>(in, tgt, n, c, ws + NB);
    hinge_final_kernel<<<1, 256, 0, stream>>>(ws, NB, ws + NB, out,
                                              1.0f / (float)n);
}